// HomoGATEncoderLinearDropout2_15805479649922
// MI455X (gfx1250) — compile-verified
//
#include <hip/hip_runtime.h>
#include <hip/hip_bf16.h>
#include <math.h>

// ---------------------------------------------------------------------------
// Types for CDNA5 WMMA
// ---------------------------------------------------------------------------
typedef float v2f __attribute__((ext_vector_type(2)));
typedef float v8f __attribute__((ext_vector_type(8)));

#if defined(__has_builtin)
#if __has_builtin(__builtin_amdgcn_wmma_f32_16x16x4_f32)
#define USE_WMMA_F32 1
#endif
#endif
#ifndef USE_WMMA_F32
#define USE_WMMA_F32 0
#endif

// ---------------------------------------------------------------------------
// Helpers
// ---------------------------------------------------------------------------
__device__ __forceinline__ void edge_sd(const int* __restrict__ ei, int E, int e,
                                        int& s, int& d) {
  if (e < E) { s = ei[e]; d = ei[E + e]; }
  else       { s = e - E; d = e - E; }          // self loops appended
}

// Monotonic-int trick for float atomic max (init with -inf).
__device__ __forceinline__ void atomicMaxF(float* addr, float v) {
  int vi = __float_as_int(v);
  if (vi >= 0) atomicMax((int*)addr, vi);
  else         atomicMin((unsigned int*)addr, (unsigned int)vi);
}

__device__ __forceinline__ float eluf(float v) {
  return v > 0.f ? v : (expf(v) - 1.f);
}

// ---------------------------------------------------------------------------
// fp32 WMMA GEMM:  C[M,Nc] = A[M,K] @ B[K,Nc]  (+bias, +elu)
// One wave computes a 16x32 C tile: one A fragment feeds TWO independent
// V_WMMA_F32_16X16X4_F32 accumulator chains (2 N-tiles).
// Requires: M % 16 == 0, K % 4 == 0, Nc % 32 == 0.
// ---------------------------------------------------------------------------
__global__ __launch_bounds__(256) void gemm_wmma_kernel(
    const float* __restrict__ A, const float* __restrict__ B,
    float* __restrict__ C, int M, int K, int Nc,
    const float* __restrict__ bias, int act) {
  int wave = (int)((blockIdx.x * (unsigned)blockDim.x + threadIdx.x) >> 5);
  int lane = threadIdx.x & 31;
  int tn2     = Nc >> 5;                     // pairs of 16-wide N tiles
  int tiles_m = M >> 4;
  int tm = wave / tn2;
  int tp = wave - tm * tn2;
  if (tm >= tiles_m) return;                 // wave-uniform exit

  int half = lane >> 4;                      // 0: lanes 0-15, 1: lanes 16-31
  int l16  = lane & 15;
  int mrow  = tm * 16 + l16;                 // A row this lane supplies
  int ncol0 = tp * 32 + l16;                 // first C column
  int ncol1 = ncol0 + 16;                    // second C column

#if USE_WMMA_F32
  v8f c0 = {0.f, 0.f, 0.f, 0.f, 0.f, 0.f, 0.f, 0.f};
  v8f c1 = c0;
  // A 16x4 frag: lane half 0 holds K=k..k+1, half 1 holds K=k+2..k+3.
  const float* Ap = A + (size_t)mrow * K + 2 * half;          // 8B aligned
  // B 4x16 frag: row k+2*half (+1 in .y), column = lane's ncol.
  const float* Bp = B + (size_t)(2 * half) * Nc;
  size_t bstep = (size_t)4 * Nc;
  for (int k = 0; k < K; k += 4) {
    v2f a = *(const v2f*)Ap;                                   // global_load_b64
    v2f b0, b1;
    b0.x = Bp[ncol0];      b0.y = Bp[Nc + ncol0];
    b1.x = Bp[ncol1];      b1.y = Bp[Nc + ncol1];
    c0 = __builtin_amdgcn_wmma_f32_16x16x4_f32(false, a, false, b0,
                                               (short)0, c0, false, false);
    c1 = __builtin_amdgcn_wmma_f32_16x16x4_f32(false, a, false, b1,
                                               (short)0, c1, false, false);
    Ap += 4;
    Bp += bstep;
  }
  float bv0 = bias ? bias[ncol0] : 0.f;
  float bv1 = bias ? bias[ncol1] : 0.f;
  float* Crow = C + (size_t)(tm * 16 + 8 * half) * Nc;
#pragma unroll
  for (int i = 0; i < 8; ++i) {
    // C layout: vgpr i -> row i (lanes 0-15), row i+8 (lanes 16-31)
    float v0 = c0[i] + bv0;
    float v1 = c1[i] + bv1;
    if (act) { v0 = eluf(v0); v1 = eluf(v1); }
    Crow[(size_t)i * Nc + ncol0] = v0;
    Crow[(size_t)i * Nc + ncol1] = v1;
  }
#else
  // Scalar fallback (host pass / missing builtin): identical mapping.
  float bv0 = bias ? bias[ncol0] : 0.f;
  float bv1 = bias ? bias[ncol1] : 0.f;
#pragma unroll 1
  for (int i = 0; i < 8; ++i) {
    int r = tm * 16 + i + 8 * half;
    float a0 = 0.f, a1 = 0.f;
    const float* Ar = A + (size_t)r * K;
    for (int k = 0; k < K; ++k) {
      a0 = fmaf(Ar[k], B[(size_t)k * Nc + ncol0], a0);
      a1 = fmaf(Ar[k], B[(size_t)k * Nc + ncol1], a1);
    }
    float v0 = a0 + bv0, v1 = a1 + bv1;
    if (act) { v0 = eluf(v0); v1 = eluf(v1); }
    C[(size_t)r * Nc + ncol0] = v0;
    C[(size_t)r * Nc + ncol1] = v1;
  }
#endif
}

// ---------------------------------------------------------------------------
// Edge pass A: score[e,h] = att[h] . leakyrelu(xl[src] + xr[dst]) ; atomic max
// One wave per edge. C (per head) == 64, F = H*64.
// ---------------------------------------------------------------------------
__global__ __launch_bounds__(256) void edge_score_kernel(
    const float* __restrict__ xl, const float* __restrict__ xr,
    const float* __restrict__ att, const int* __restrict__ ei,
    int E, int Etot, int H,
    float* __restrict__ score, float* __restrict__ m) {
  int gw = (int)((blockIdx.x * (unsigned)blockDim.x + threadIdx.x) >> 5);
  int lane = threadIdx.x & 31;
  if (gw >= Etot) return;
  int s, d;
  edge_sd(ei, E, gw, s, d);
  int F = H << 6;
  const float* xls = xl + (size_t)s * F;
  const float* xrd = xr + (size_t)d * F;
  for (int h = 0; h < H; ++h) {
    int b = h << 6;
    float a0 = xls[b + lane]      + xrd[b + lane];
    float a1 = xls[b + 32 + lane] + xrd[b + 32 + lane];
    a0 = a0 > 0.f ? a0 : 0.2f * a0;
    a1 = a1 > 0.f ? a1 : 0.2f * a1;
    float acc = att[b + lane] * a0 + att[b + 32 + lane] * a1;
#pragma unroll
    for (int off = 16; off; off >>= 1) acc += __shfl_xor(acc, off, 32);
    if (lane == 0) {
      score[(size_t)gw * H + h] = acc;
      atomicMaxF(&m[(size_t)d * H + h], acc);
    }
  }
}

// ---------------------------------------------------------------------------
// Edge pass B: a = exp(score - m[dst]); denom[dst] += a ; score <- a
// One thread per (edge, head).
// ---------------------------------------------------------------------------
__global__ __launch_bounds__(256) void edge_exp_kernel(
    const int* __restrict__ ei, int E, int Etot, int H,
    float* __restrict__ score, const float* __restrict__ m,
    float* __restrict__ den) {
  long long i = (long long)blockIdx.x * blockDim.x + threadIdx.x;
  long long tot = (long long)Etot * H;
  if (i >= tot) return;
  int e = (int)(i / H);
  int h = (int)(i - (long long)e * H);
  int s, d;
  edge_sd(ei, E, e, s, d);
  (void)s;
  float a = expf(score[i] - m[(size_t)d * H + h]);
  score[i] = a;
  atomicAdd(&den[(size_t)d * H + h], a);
}

// ---------------------------------------------------------------------------
// Edge pass C: out[dst] += xl[src] * (a / denom[dst])   (per head)
// One wave per edge; lane strides channels by 32; head = channel >> 6.
// ---------------------------------------------------------------------------
__global__ __launch_bounds__(256) void edge_aggr_kernel(
    const float* __restrict__ xl, const int* __restrict__ ei,
    int E, int Etot, int H,
    const float* __restrict__ score, const float* __restrict__ den,
    float* __restrict__ out) {
  int gw = (int)((blockIdx.x * (unsigned)blockDim.x + threadIdx.x) >> 5);
  int lane = threadIdx.x & 31;
  if (gw >= Etot) return;
  int s, d;
  edge_sd(ei, E, gw, s, d);
  int F = H << 6;
  const float* xls = xl + (size_t)s * F;
  float* outd = out + (size_t)d * F;
  int T = F >> 5;
  for (int t = 0; t < T; ++t) {
    int i = (t << 5) + lane;
    int h = i >> 6;
    float alpha = score[(size_t)gw * H + h] / den[(size_t)d * H + h];
    atomicAdd(&outd[i], xls[i] * alpha);
  }
}

// ---------------------------------------------------------------------------
// Node-wise: x <- elu(x + bias[i % F])
// ---------------------------------------------------------------------------
__global__ __launch_bounds__(256) void bias_elu_kernel(
    float* __restrict__ x, const float* __restrict__ bias, int F, long long n) {
  long long i = (long long)blockIdx.x * blockDim.x + threadIdx.x;
  if (i >= n) return;
  float v = x[i] + bias[(int)(i % F)];
  x[i] = eluf(v);
}

__global__ __launch_bounds__(256) void fill_kernel(float* __restrict__ p,
                                                   long long n, float v) {
  long long i = (long long)blockIdx.x * blockDim.x + threadIdx.x;
  if (i < n) p[i] = v;
}

// ---------------------------------------------------------------------------
// Host-side launch helpers
// ---------------------------------------------------------------------------
static inline void launch_gemm(const float* A, const float* B, float* C,
                               int M, int K, int Nc, const float* bias, int act,
                               hipStream_t stream) {
  long long waves = (long long)(M / 16) * (Nc / 32);   // 16x32 C tile per wave
  int blocks = (int)((waves + 7) / 8);                 // 8 waves per 256-thread block
  gemm_wmma_kernel<<<blocks, 256, 0, stream>>>(A, B, C, M, K, Nc, bias, act);
}

static inline void launch_fill(float* p, long long n, float v, hipStream_t stream) {
  int blocks = (int)((n + 255) / 256);
  fill_kernel<<<blocks, 256, 0, stream>>>(p, n, v);
}

extern "C" void kernel_launch(void* const* d_in, const int* in_sizes, int n_in,
                              void* d_out, int out_size, void* d_ws, size_t ws_size,
                              hipStream_t stream) {
  const float* x    = (const float*)d_in[0];
  const int*   ei   = (const int*)d_in[1];
  const float* W1l  = (const float*)d_in[2];
  const float* W1r  = (const float*)d_in[3];
  const float* att1 = (const float*)d_in[4];
  const float* b1   = (const float*)d_in[5];
  const float* W2l  = (const float*)d_in[6];
  const float* W2r  = (const float*)d_in[7];
  const float* att2 = (const float*)d_in[8];
  const float* b2   = (const float*)d_in[9];
  const float* Wlin = (const float*)d_in[10];
  const float* blin = (const float*)d_in[11];
  float* out = (float*)d_out;

  const int D = 128, F1 = 512, H1 = 8, C2 = 64, OUTC = 32;
  int N    = in_sizes[0] / D;        // 20000
  int E    = in_sizes[1] / 2;        // 320000
  int Etot = E + N;                  // + self loops

  // ---- workspace layout (floats) ----
  float* ws = (float*)d_ws;
  size_t NF1 = (size_t)N * F1;
  float* xl1    = ws;                               // [N,512]
  float* xr1    = ws + NF1;                         // [N,512]
  float* out1   = ws + 2 * NF1;                     // [N,512] -> h1 after elu
  float* score1 = ws + 3 * NF1;                     // [Etot,8]
  float* m1     = score1 + (size_t)Etot * H1;       // [N,8]
  float* den1   = m1 + (size_t)N * H1;              // [N,8]
  // layer-2 buffers reuse the (now dead) xl1/xr1 region
  size_t NC2 = (size_t)N * C2;
  float* xl2    = ws;                               // [N,64]
  float* xr2    = ws + NC2;                         // [N,64]
  float* out2   = ws + 2 * NC2;                     // [N,64] -> h2 after elu
  float* score2 = ws + 3 * NC2;                     // [Etot]
  float* m2     = score2 + Etot;                    // [N]
  float* den2   = m2 + N;                           // [N]
  (void)ws_size; (void)n_in; (void)out_size;

  int eblocks  = (Etot + 7) / 8;                    // wave-per-edge kernels

  // ================= Layer 1 =================
  launch_gemm(x, W1l, xl1, N, D, F1, nullptr, 0, stream);
  launch_gemm(x, W1r, xr1, N, D, F1, nullptr, 0, stream);
  launch_fill(m1,   (long long)N * H1, -INFINITY, stream);
  launch_fill(den1, (long long)N * H1, 0.f, stream);
  launch_fill(out1, (long long)NF1,    0.f, stream);

  edge_score_kernel<<<eblocks, 256, 0, stream>>>(xl1, xr1, att1, ei, E, Etot, H1,
                                                 score1, m1);
  {
    long long tot = (long long)Etot * H1;
    edge_exp_kernel<<<(int)((tot + 255) / 256), 256, 0, stream>>>(ei, E, Etot, H1,
                                                                  score1, m1, den1);
  }
  edge_aggr_kernel<<<eblocks, 256, 0, stream>>>(xl1, ei, E, Etot, H1,
                                                score1, den1, out1);
  bias_elu_kernel<<<(int)((NF1 + 255) / 256), 256, 0, stream>>>(out1, b1, F1,
                                                                (long long)NF1);

  // ================= Layer 2 (H=1, C=64, mean over 1 head == identity) ======
  launch_gemm(out1, W2l, xl2, N, F1, C2, nullptr, 0, stream);
  launch_gemm(out1, W2r, xr2, N, F1, C2, nullptr, 0, stream);
  launch_fill(m2,   N, -INFINITY, stream);
  launch_fill(den2, N, 0.f, stream);
  launch_fill(out2, (long long)NC2, 0.f, stream);

  edge_score_kernel<<<eblocks, 256, 0, stream>>>(xl2, xr2, att2, ei, E, Etot, 1,
                                                 score2, m2);
  edge_exp_kernel<<<(Etot + 255) / 256, 256, 0, stream>>>(ei, E, Etot, 1,
                                                          score2, m2, den2);
  edge_aggr_kernel<<<eblocks, 256, 0, stream>>>(xl2, ei, E, Etot, 1,
                                                score2, den2, out2);
  bias_elu_kernel<<<(int)((NC2 + 255) / 256), 256, 0, stream>>>(out2, b2, C2,
                                                                (long long)NC2);

  // ================= Final linear + ELU =================
  launch_gemm(out2, Wlin, out, N, C2, OUTC, blin, 1, stream);
}